// GCNLayer_15375982920182
// MI455X (gfx1250) — compile-verified
//
#include <hip/hip_runtime.h>
#include <hip/hip_bf16.h>
#include <cstddef>
#include <cstdint>

// GCN aggregation, CSR-by-destination formulation, gfx1250-tuned.
// d_in[0]=u_f [N_U,64] f32, d_in[1]=v_f [N_V,64] f32, d_in[2]=src [E] i32, d_in[3]=dst [E] i32
// d_out = [N_U+N_V, 64] f32

// ---- gfx1250 async global->LDS staging (guarded; falls back to plain loads) ----
#if defined(__gfx1250__) && __has_builtin(__builtin_amdgcn_global_load_async_to_lds_b32)
#define USE_ASYNC_LDS 1
#endif

#ifdef USE_ASYNC_LDS
typedef __attribute__((address_space(1))) int g_int;   // global (AS1)
typedef __attribute__((address_space(3))) int l_int;   // LDS (AS3)
__device__ __forceinline__ void wait_async0() {
#if __has_builtin(__builtin_amdgcn_s_wait_asynccnt)
  __builtin_amdgcn_s_wait_asynccnt(0);
  asm volatile("" ::: "memory");
#else
  asm volatile("s_wait_asynccnt 0x0" ::: "memory");
#endif
}
#endif

#define WPB 8      // waves per 256-thread block
#define CHUNK 64   // edge-index staging chunk per wave (bytes: 256)

// ---------------- degree counting (edge-parallel, int atomics) ----------------
__global__ void k_degrees(const int* __restrict__ src, const int* __restrict__ dst,
                          int* __restrict__ out_deg, int* __restrict__ in_deg, int E) {
  int e = blockIdx.x * blockDim.x + threadIdx.x;
  if (e < E) {
    atomicAdd(&out_deg[src[e]], 1);
    atomicAdd(&in_deg[dst[e]], 1);
  }
}

// ---------------- row offsets: LDS block scan + one atomic per block ----------
// (order of blocks is irrelevant: each dst node just needs a contiguous segment)
__global__ void k_offsets(const int* __restrict__ out_deg, const int* __restrict__ in_deg,
                          int* __restrict__ row_off, float* __restrict__ rs_out,
                          int* __restrict__ counter, int NTOT) {
  __shared__ int sdata[256];
  __shared__ int sbase;
  int t = threadIdx.x;
  int n = blockIdx.x * blockDim.x + t;
  int val = (n < NTOT) ? in_deg[n] : 0;
  sdata[t] = val;
  __syncthreads();
  for (int o = 1; o < 256; o <<= 1) {  // Hillis-Steele inclusive scan
    int x = (t >= o) ? sdata[t - o] : 0;
    __syncthreads();
    sdata[t] += x;
    __syncthreads();
  }
  int incl = sdata[t];
  if (t == 255) sbase = atomicAdd(counter, incl);  // block total -> global base
  __syncthreads();
  if (n < NTOT) {
    row_off[n] = sbase + incl - val;  // exclusive offset
    int od = out_deg[n];
    rs_out[n] = rsqrtf((float)(od < 1 ? 1 : od));
  }
}

// ---------------- bucket fill: edge -> CSR slot --------------------------------
__global__ void k_fill(const int* __restrict__ src, const int* __restrict__ dst,
                       const int* __restrict__ row_off, int* __restrict__ cursor,
                       int* __restrict__ edge_src, int E) {
  int e = blockIdx.x * blockDim.x + threadIdx.x;
  if (e < E) {
    int d = dst[e];
    int pos = row_off[d] + atomicAdd(&cursor[d], 1);
    edge_src[pos] = src[e];
  }
}

// ---------------- pre-scale features by rsqrt(out_deg), fusing concat ---------
__global__ void k_prescale(const float* __restrict__ u, const float* __restrict__ v,
                           const float* __restrict__ rs_out, float* __restrict__ featS,
                           int N_U, int NTOT) {
  int idx = blockIdx.x * blockDim.x + threadIdx.x;  // one float4 per thread
  if (idx >= NTOT * 16) return;
  int node = idx >> 4;
  int part = idx & 15;
  const float4* sp = (node < N_U)
      ? ((const float4*)u) + (size_t)node * 16 + part
      : ((const float4*)v) + (size_t)(node - N_U) * 16 + part;
  float s = rs_out[node];
  float4 val = *sp;
  val.x *= s; val.y *= s; val.z *= s; val.w *= s;
  ((float4*)featS)[(size_t)node * 16 + part] = val;
}

// ---------------- aggregation: one wave32 per destination node ----------------
// Edge-index chunk is staged into LDS via gfx1250 async global->LDS copies
// (ASYNCcnt), then consumed as uniform broadcast reads. Each lane owns a
// float2 column slice: wave reads a full 256B row per edge (coalesced,
// L2-resident), accumulates in registers, applies rsqrt(in_deg), one store.
// global_prefetch_b8 hides L2 gather latency of the next row.
template <bool PRE>
__global__ void k_aggregate(const float* __restrict__ featS,
                            const float* __restrict__ u, const float* __restrict__ v,
                            const float* __restrict__ rs_out,
                            const int* __restrict__ edge_src,
                            const int* __restrict__ row_off,
                            const int* __restrict__ in_deg,
                            float* __restrict__ out, int N_U, int NTOT) {
  __shared__ int s_idx[WPB][CHUNK];
  int wid  = threadIdx.x >> 5;
  int lane = threadIdx.x & 31;
  int node = blockIdx.x * WPB + wid;
  if (node >= NTOT) return;
  int beg = row_off[node];
  int cnt = in_deg[node];
  float ax = 0.0f, ay = 0.0f;

  for (int base = 0; base < cnt; base += CHUNK) {
    int n = cnt - base;
    if (n > CHUNK) n = CHUNK;
#ifdef USE_ASYNC_LDS
    for (int k = lane; k < n; k += 32) {
      __builtin_amdgcn_global_load_async_to_lds_b32(
          (g_int*)(&edge_src[beg + base + k]),
          (l_int*)(&s_idx[wid][k]), 0, 0);
    }
    wait_async0();
#else
    for (int k = lane; k < n; k += 32) {
      s_idx[wid][k] = edge_src[beg + base + k];
    }
#endif
    for (int j = 0; j < n; ++j) {
      int s = s_idx[wid][j];
      if (j + 1 < n) {  // prefetch next gathered row (global_prefetch_b8)
        int sn = s_idx[wid][j + 1];
        const float* pn;
        if (PRE) pn = featS + (size_t)sn * 64;
        else     pn = (sn < N_U) ? u + (size_t)sn * 64 : v + (size_t)(sn - N_U) * 64;
        __builtin_prefetch(pn + lane * 2, 0, 3);
      }
      const float* row;
      float sc;
      if (PRE) { row = featS + (size_t)s * 64; sc = 1.0f; }
      else {
        row = (s < N_U) ? u + (size_t)s * 64 : v + (size_t)(s - N_U) * 64;
        sc = rs_out[s];
      }
      float2 val = *(const float2*)(row + lane * 2);
      ax = fmaf(val.x, sc, ax);
      ay = fmaf(val.y, sc, ay);
    }
  }
  float rin = rsqrtf((float)(cnt < 1 ? 1 : cnt));
  float2 res;
  res.x = ax * rin;
  res.y = ay * rin;
  *(float2*)(out + (size_t)node * 64 + lane * 2) = res;
}

// ---------------- fallback: edge-parallel atomic scatter (tiny workspace) -----
__global__ void k_scatter_atomic(const float* __restrict__ u, const float* __restrict__ v,
                                 const int* __restrict__ src, const int* __restrict__ dst,
                                 const int* __restrict__ out_deg, const int* __restrict__ in_deg,
                                 float* __restrict__ out, int N_U, int E) {
  int e = blockIdx.x * (blockDim.x >> 5) + (threadIdx.x >> 5);
  int lane = threadIdx.x & 31;
  if (e >= E) return;
  int s = src[e], d = dst[e];
  int od = out_deg[s]; if (od < 1) od = 1;
  int id = in_deg[d];  if (id < 1) id = 1;
  float sc = rsqrtf((float)od) * rsqrtf((float)id);
  const float* row = (s < N_U) ? u + (size_t)s * 64 : v + (size_t)(s - N_U) * 64;
  float2 val = *(const float2*)(row + lane * 2);
  float* o = out + (size_t)d * 64 + lane * 2;
  atomicAdd(o + 0, val.x * sc);
  atomicAdd(o + 1, val.y * sc);
}

extern "C" void kernel_launch(void* const* d_in, const int* in_sizes, int n_in,
                              void* d_out, int out_size, void* d_ws, size_t ws_size,
                              hipStream_t stream) {
  const float* u   = (const float*)d_in[0];
  const float* v   = (const float*)d_in[1];
  const int*   src = (const int*)d_in[2];
  const int*   dst = (const int*)d_in[3];
  const int D    = 64;
  const int N_U  = in_sizes[0] / D;
  const int N_V  = in_sizes[1] / D;
  const int NTOT = N_U + N_V;
  const int E    = in_sizes[2];
  float* out = (float*)d_out;

  // workspace layout
  char* ws = (char*)d_ws;
  size_t off = 0;
  int*   out_deg = (int*)(ws + off);   off += (size_t)NTOT * 4;
  int*   in_deg  = (int*)(ws + off);   off += (size_t)NTOT * 4;
  int*   cursor  = (int*)(ws + off);   off += (size_t)NTOT * 4;
  int*   row_off = (int*)(ws + off);   off += (size_t)NTOT * 4;
  float* rs_out  = (float*)(ws + off); off += (size_t)NTOT * 4;
  int*   counter = (int*)(ws + off);   off += 256;
  int*   edge_src = (int*)(ws + off);  off += (size_t)E * 4;
  off = (off + 15) & ~(size_t)15;      // align for float4
  size_t csrBytes = off;
  float* featS = (float*)(ws + off);
  size_t fullBytes = off + (size_t)NTOT * D * 4;

  const int B = 256;
  dim3 blk(B);
  int gridE = (E + B - 1) / B;
  int gridN = (NTOT + B - 1) / B;
  int gridNodeW = (NTOT + WPB - 1) / WPB;

  if (ws_size >= csrBytes) {
    // zero degrees/cursor/counter region (row_off/rs_out get overwritten anyway)
    (void)hipMemsetAsync(ws, 0, (size_t)NTOT * 20 + 256, stream);
    k_degrees<<<gridE, blk, 0, stream>>>(src, dst, out_deg, in_deg, E);
    k_offsets<<<gridN, blk, 0, stream>>>(out_deg, in_deg, row_off, rs_out, counter, NTOT);
    k_fill<<<gridE, blk, 0, stream>>>(src, dst, row_off, cursor, edge_src, E);
    if (ws_size >= fullBytes) {
      int gridP = (NTOT * 16 + B - 1) / B;
      k_prescale<<<gridP, blk, 0, stream>>>(u, v, rs_out, featS, N_U, NTOT);
      k_aggregate<true><<<gridNodeW, blk, 0, stream>>>(featS, u, v, rs_out, edge_src,
                                                       row_off, in_deg, out, N_U, NTOT);
    } else {
      k_aggregate<false><<<gridNodeW, blk, 0, stream>>>(nullptr, u, v, rs_out, edge_src,
                                                        row_off, in_deg, out, N_U, NTOT);
    }
  } else {
    // minimal-workspace fallback: atomic scatter
    (void)hipMemsetAsync(ws, 0, (size_t)NTOT * 8, stream);
    (void)hipMemsetAsync(out, 0, (size_t)out_size * 4, stream);
    k_degrees<<<gridE, blk, 0, stream>>>(src, dst, out_deg, in_deg, E);
    int gridEw = (E + (B / 32) - 1) / (B / 32);
    k_scatter_atomic<<<gridEw, blk, 0, stream>>>(u, v, src, dst, out_deg, in_deg, out, N_U, E);
  }
}